// SparseMaxPooling_56676388438422
// MI455X (gfx1250) — compile-verified
//
#include <hip/hip_runtime.h>
#include <hip/hip_bf16.h>

// Sparse max pooling over a 3^3 kernel neighborhood.
//   in_feat      : [N_IN, 96]  float32
//   neighbor_map : [N_OUT, 27] int (row index into in_feat per kernel offset)
//   out          : [N_OUT, 96] float32 = max over the 27 gathered rows
//
// MI455X reasoning: in_feat (153.6 MB) fits in the 192 MB global L2, so this
// is an L2-bandwidth-bound gather (~1.04 GB of L2 reads, ~204 MB HBM).
// One wave32 per output row; lane l owns channels [3l, 3l+3). Gather bases
// are broadcast to SGPRs via v_readlane so every neighbor load is a single
// saddr+voffset global_load_b96 covering 384 contiguous bytes per wave.
// global_prefetch_b8 warms all 81 cachelines of the row's gather set up front.

#define KVOL 27
#define CCH  96
#define WAVES_PER_BLOCK 8
#define BLOCK_THREADS (WAVES_PER_BLOCK * 32)

struct __attribute__((aligned(4))) Float3 {
    float x, y, z;
};

__global__ __launch_bounds__(BLOCK_THREADS)
void SparseMaxPooling_kernel(const float* __restrict__ in_feat,
                             const int* __restrict__ nmap,
                             float* __restrict__ out,
                             int n_out) {
    const int lane = threadIdx.x & 31;
    const int wave = threadIdx.x >> 5;
    const int row  = blockIdx.x * WAVES_PER_BLOCK + wave;
    if (row >= n_out) return;   // row is wave-uniform: whole wave exits together

    // Lane-parallel load of this row's 27 neighbor indices. Clamp instead of
    // branching so there is no EXEC save/restore (lanes 27..31 just reload
    // index 26; they are never read back via v_readlane).
    const int klane = lane < (KVOL - 1) ? lane : (KVOL - 1);
    const int idx_lane = nmap[row * KVOL + klane];

    // Prefetch every gathered row (96 floats = 384 B = 3 cachelines each).
    // Emits global_prefetch_b8; kicks all 27 gathers off before the max chain.
    {
        const char* p = (const char*)in_feat + (size_t)idx_lane * (CCH * sizeof(float));
        __builtin_prefetch(p,       0, 3);
        __builtin_prefetch(p + 128, 0, 3);
        __builtin_prefetch(p + 256, 0, 3);
    }

    const int coff = lane * 3;  // this lane's first channel

    // Neighbor 0 initializes the accumulators (KVOL >= 1 always).
    const int idx0 = __builtin_amdgcn_readlane(idx_lane, 0);
    Float3 v0 = *(const Float3*)(in_feat + (size_t)idx0 * CCH + coff);
    float a0 = v0.x, a1 = v0.y, a2 = v0.z;

#pragma unroll
    for (int k = 1; k < KVOL; ++k) {
        // Broadcast index k to an SGPR -> scalar gather base, saddr addressing.
        const int idx = __builtin_amdgcn_readlane(idx_lane, k);
        const Float3 v = *(const Float3*)(in_feat + (size_t)idx * CCH + coff);
        a0 = fmaxf(a0, v.x);
        a1 = fmaxf(a1, v.y);
        a2 = fmaxf(a2, v.z);
    }

    Float3 r;
    r.x = a0; r.y = a1; r.z = a2;
    *(Float3*)(out + (size_t)row * CCH + coff) = r;   // coalesced 384 B/wave
}

extern "C" void kernel_launch(void* const* d_in, const int* in_sizes, int n_in,
                              void* d_out, int out_size, void* d_ws, size_t ws_size,
                              hipStream_t stream) {
    const float* in_feat = (const float*)d_in[0];
    const int*   nmap    = (const int*)d_in[1];   // integer input -> const int*
    float*       out     = (float*)d_out;

    const int n_out  = in_sizes[1] / KVOL;        // neighbor_map has n_out*27 elems
    const int blocks = (n_out + WAVES_PER_BLOCK - 1) / WAVES_PER_BLOCK;

    SparseMaxPooling_kernel<<<blocks, BLOCK_THREADS, 0, stream>>>(in_feat, nmap, out, n_out);
}